// GatedTreeLSTMCell_66571993088454
// MI455X (gfx1250) — compile-verified
//
#include <hip/hip_runtime.h>
#include <hip/hip_bf16.h>

// GatedTreeLSTM forward for MI455X (gfx1250), fp32 via V_WMMA_F32_16X16X4_F32,
// with async global->LDS staging (ASYNCcnt) for the recurrent A-operand.
// B=64, T=256, I=H=1024, 5H=5120.

typedef float v2f __attribute__((ext_vector_type(2)));
typedef float v8f __attribute__((ext_vector_type(8)));

#define B_ 64
#define T_ 256
#define I_ 1024
#define H_ 1024
#define G5 5120
#define SLOTS 257  // T+1

static __device__ __forceinline__ v8f wmma_f32(v2f a, v2f b, v8f c) {
  // D = A(16x4 f32) x B(4x16 f32) + C(16x16 f32), full fp32 matrix pipe.
  return __builtin_amdgcn_wmma_f32_16x16x4_f32(false, a, false, b, (short)0, c,
                                               false, false);
}

static __device__ __forceinline__ void wait_async0() {
#if __has_builtin(__builtin_amdgcn_s_wait_asynccnt)
  __builtin_amdgcn_s_wait_asynccnt(0);
#else
  asm volatile("s_wait_asynccnt 0" ::: "memory");
#endif
}

// ---------------------------------------------------------------------------
// Zero recurrent state: ybuf0|ybuf1|cbuf (contiguous), buf slot 0, stack, sp.
// ---------------------------------------------------------------------------
__global__ __launch_bounds__(256) void init_kernel(float* __restrict__ zreg,
                                                   float* __restrict__ buf,
                                                   int* __restrict__ stk,
                                                   int* __restrict__ sp) {
  const int idx = blockIdx.x * 256 + threadIdx.x;
  if (idx < 3 * B_ * H_) zreg[idx] = 0.0f;           // ybuf0, ybuf1, cbuf
  if (idx < B_ * H_) {                               // buf[:, 0, :] = 0
    const int b = idx >> 10, h = idx & (H_ - 1);
    buf[(size_t)b * SLOTS * H_ + h] = 0.0f;
  }
  if (idx < B_ * SLOTS) stk[idx] = 0;
  if (idx < 2 * B_) sp[idx] = 0;
}

// ---------------------------------------------------------------------------
// Zx = X @ W_ih^T + bias   :  [16384,1024] x [1024,5120] -> [16384,5120]
// One wave computes a 16(M) x 64(N) strip: 4 independent WMMA chains.
// ---------------------------------------------------------------------------
__global__ __launch_bounds__(256) void zx_gemm_kernel(
    const float* __restrict__ X, const float* __restrict__ Wih,
    const float* __restrict__ bias, float* __restrict__ Zx) {
  const int lane = threadIdx.x & 31;
  const int wid  = blockIdx.x * 8 + (threadIdx.x >> 5);  // 81920 waves
  const int mTile = wid / 80;          // 1024 M-tiles
  const int nGrp  = wid - mTile * 80;  // 80 N-groups of 64 cols
  const int nl   = lane & 15;
  const int half = lane >> 4;
  const int row0 = mTile << 4;
  const int col0 = nGrp << 6;

  // A lane: row = row0+nl, K-pair at k + 2*half    (16x4 f32 A layout)
  // B lane: col = colX+nl, K-pair at k + 2*half    (4x16 f32 B layout)
  const float* ap = X + (size_t)(row0 + nl) * I_ + (half << 1);
  const float* bp = Wih + (size_t)(col0 + nl) * I_ + (half << 1);

  v8f c0 = {}, c1 = {}, c2 = {}, c3 = {};
#pragma unroll 4
  for (int k = 0; k < I_; k += 4) {
    v2f a  = *(const v2f*)(ap + k);
    v2f b0 = *(const v2f*)(bp + k);
    v2f b1 = *(const v2f*)(bp + 16 * I_ + k);
    v2f b2 = *(const v2f*)(bp + 32 * I_ + k);
    v2f b3 = *(const v2f*)(bp + 48 * I_ + k);
    c0 = wmma_f32(a, b0, c0);
    c1 = wmma_f32(a, b1, c1);
    c2 = wmma_f32(a, b2, c2);
    c3 = wmma_f32(a, b3, c3);
  }

  const float bb0 = bias[col0 + nl];
  const float bb1 = bias[col0 + 16 + nl];
  const float bb2 = bias[col0 + 32 + nl];
  const float bb3 = bias[col0 + 48 + nl];
#pragma unroll
  for (int v = 0; v < 8; ++v) {
    const int row = row0 + v + (half << 3);  // C layout: VGPR v -> M=v(+8)
    const size_t base = (size_t)row * G5 + col0 + nl;
    Zx[base]      = c0[v] + bb0;
    Zx[base + 16] = c1[v] + bb1;
    Zx[base + 32] = c2[v] + bb2;
    Zx[base + 48] = c3[v] + bb3;
  }
}

// ---------------------------------------------------------------------------
// One recurrent timestep, fully fused:
//   y tile async-DMA'd to LDS (double buffered, ASYNCcnt) ->
//   acts_hh = y_in @ W_hh^T (WMMA, A from LDS) -> LDS ->
//   gates + stack/retain + c,y update.
// Grid: 256 blocks = (n: 64 h-tiles) x (m: 4 batch-tiles); block = 5 waves,
// wave g owns gate g's 16x16 tile.
// ---------------------------------------------------------------------------
__global__ __launch_bounds__(160) void step_kernel(
    const float* __restrict__ Zx, const float* __restrict__ Whh,
    const float* __restrict__ y_in, float* __restrict__ y_out,
    float* __restrict__ cbuf, float* __restrict__ buf, int* __restrict__ stk,
    const int* __restrict__ sp_in, int* __restrict__ sp_out,
    const int* __restrict__ pp_all, float* __restrict__ out, int t) {
  __shared__ float yA[2][16][258];    // double-buffered y K-chunk (pad: align+banks)
  __shared__ float sacts[5][16][17];  // [gate][b-in-tile][h-in-tile] (+pad)

  const int n = blockIdx.x >> 2;   // h-tile
  const int m = blockIdx.x & 3;    // batch-tile
  const int g = threadIdx.x >> 5;  // gate (wave id)
  const int lane = threadIdx.x & 31;
  const int nl = lane & 15;
  const int half = lane >> 4;

  const float* bp = Whh + (size_t)(g * H_ + (n << 4) + nl) * H_ + (half << 1);

  // Stage one 16x256 K-chunk of the y batch-tile into LDS asynchronously.
  // Threads 0..127 issue 8 b128 transfers each (uniform per-wave ASYNCcnt).
  auto stage = [&](int kc, int sbuf) {
    if (threadIdx.x < 128) {
#pragma unroll
      for (int j = 0; j < 8; ++j) {
        const int idx = threadIdx.x + j * 128;  // 0..1023 float4 slots
        const int row = idx >> 6;               // 16 rows x 64 float4
        const int c4  = idx & 63;
        const float* gp =
            y_in + (size_t)((m << 4) + row) * H_ + kc + (c4 << 2);
        // LDS aperture: low 32 bits of the generic pointer == LDS byte offset
        const unsigned lo = (unsigned)(size_t)&yA[sbuf][row][c4 << 2];
        asm volatile("global_load_async_to_lds_b128 %0, %1, off" ::"v"(lo),
                     "v"((unsigned long long)(size_t)gp)
                     : "memory");
      }
    }
  };

  v8f c0 = {}, c1 = {};  // two chains to break WMMA RAW latency
  stage(0, 0);
  wait_async0();
  __syncthreads();
#pragma unroll
  for (int ch = 0; ch < 4; ++ch) {
    if (ch < 3) stage((ch + 1) << 8, (ch + 1) & 1);  // overlap DMA w/ WMMA
    const float* arow = &yA[ch & 1][nl][half << 1];  // LDS A fragments
    const float* bpc = bp + (ch << 8);
#pragma unroll 4
    for (int k = 0; k < 256; k += 8) {
      v2f a0 = *(const v2f*)(arow + k);
      v2f b0 = *(const v2f*)(bpc + k);
      v2f a1 = *(const v2f*)(arow + k + 4);
      v2f b1 = *(const v2f*)(bpc + k + 4);
      c0 = wmma_f32(a0, b0, c0);
      c1 = wmma_f32(a1, b1, c1);
    }
    wait_async0();     // next chunk landed (per wave)...
    __syncthreads();   // ...and for all waves; buffer free for reuse
  }
#pragma unroll
  for (int v = 0; v < 8; ++v) sacts[g][v + (half << 3)][nl] = c0[v] + c1[v];
  __syncthreads();

  const int kslot = t + 1;
  for (int i = threadIdx.x; i < 256; i += 160) {
    const int lb = i >> 4, hl = i & 15;
    const int b = (m << 4) + lb;
    const int h = (n << 4) + hl;

    const size_t zb = ((size_t)b * T_ + t) * G5 + h;
    const float a_r = sacts[0][lb][hl] + Zx[zb];
    const float a_i = sacts[1][lb][hl] + Zx[zb + H_];
    const float a_f = sacts[2][lb][hl] + Zx[zb + 2 * H_];
    const float a_o = sacts[3][lb][hl] + Zx[zb + 3 * H_];
    const float a_u = sacts[4][lb][hl] + Zx[zb + 4 * H_];
    const float rg = 1.0f / (1.0f + __expf(-a_r));
    const float ig = 1.0f / (1.0f + __expf(-a_i));
    const float fg = 1.0f / (1.0f + __expf(-a_f));
    const float og = 1.0f / (1.0f + __expf(-a_o));
    const float up = tanhf(a_u);

    // per-batch stack scalars (redundant across h, cheap L1 hits)
    const int pp = pp_all[b * T_ + t];
    const int sp0 = sp_in[b];
    const int parent_ptr = stk[b * SLOTS + sp0];
    const bool push = pp > 0;
    int sp1 = sp0;
    if (push) sp1 = sp0 + 1;
    else if (pp < 0) { sp1 = sp0 + pp; if (sp1 < 0) sp1 = 0; }

    const float parent = buf[((size_t)b * SLOTS + parent_ptr) * H_ + h];
    const float z = fminf(1.0f, (1.0f - parent) / rg);
    const float child = parent + rg * z;
    buf[((size_t)b * SLOTS + kslot) * H_ + h] = child;

    float retain;
    if (push) {
      retain = child;  // pushed slot holds kslot; bypass the (racing) write
    } else {
      const int rp = stk[b * SLOTS + sp1];  // slot untouched this step
      retain = buf[((size_t)b * SLOTS + rp) * H_ + h];
    }

    const float c = cbuf[b * H_ + h];
    const float cand = (1.0f - fg) * c + ig * up;
    const float ct = retain * c + (1.0f - retain) * cand;
    const float yt = og * tanhf(ct);
    cbuf[b * H_ + h] = ct;
    y_out[b * H_ + h] = yt;
    out[((size_t)b * T_ + t) * H_ + h] = yt;

    // single writer per batch element commits stack/sp for next step
    if (n == 0 && hl == 0) {
      if (push) stk[b * SLOTS + sp1] = kslot;
      sp_out[b] = sp1;
    }
  }
}

// ---------------------------------------------------------------------------
extern "C" void kernel_launch(void* const* d_in, const int* in_sizes, int n_in,
                              void* d_out, int out_size, void* d_ws,
                              size_t ws_size, hipStream_t stream) {
  (void)in_sizes; (void)n_in; (void)out_size; (void)ws_size;
  const float* x       = (const float*)d_in[0];  // [B,T,I]
  const int*   pushpop = (const int*)d_in[1];    // [B,T]
  const float* Wih     = (const float*)d_in[2];  // [5H,I]
  const float* Whh     = (const float*)d_in[3];  // [5H,H]
  const float* bias    = (const float*)d_in[4];  // [5H]
  float* out = (float*)d_out;                    // [B,T,H]

  float* w     = (float*)d_ws;
  float* Zx    = w;                                   // 16384*5120
  float* buf   = Zx + (size_t)B_ * T_ * G5;           // 64*257*1024
  float* ybuf0 = buf + (size_t)B_ * SLOTS * H_;
  float* ybuf1 = ybuf0 + B_ * H_;
  float* cbuf  = ybuf1 + B_ * H_;
  int*   stk   = (int*)(cbuf + B_ * H_);              // 64*257
  int*   sp    = stk + B_ * SLOTS;                    // 2*64 (double buffer)

  init_kernel<<<768, 256, 0, stream>>>(ybuf0, buf, stk, sp);
  zx_gemm_kernel<<<10240, 256, 0, stream>>>(x, Wih, bias, Zx);

  for (int t = 0; t < T_; ++t) {
    const float* yin  = (t & 1) ? ybuf1 : ybuf0;
    float*       yout = (t & 1) ? ybuf0 : ybuf1;
    const int*   spin = sp + (t & 1) * B_;
    int*         spout = sp + ((t + 1) & 1) * B_;
    step_kernel<<<256, 160, 0, stream>>>(Zx, Whh, yin, yout, cbuf, buf, stk,
                                         spin, spout, pushpop, out, t);
  }
}